// NCWNO1d_68719476864
// MI455X (gfx1250) — compile-verified
//
#include <hip/hip_runtime.h>
#include <math.h>

#define BSZ     32
#define WID     64
#define NLEV    4
#define NSZ     1024
#define NHID    4
#define NEXP    10
#define NLIFT   32
#define NLAB    10
#define NMODES  64
#define LMAXF   20

typedef __attribute__((ext_vector_type(2))) float v2f;
typedef __attribute__((ext_vector_type(8))) float v8f;

// Daubechies db1..db10 orthonormal lowpass filters (sum = sqrt(2)), zero-padded to 20 taps.
__constant__ float DB_LO[NEXP][LMAXF] = {
  {0.70710678118654757f,0.70710678118654757f,0,0,0,0,0,0,0,0,0,0,0,0,0,0,0,0,0,0},
  {0.48296291314469025f,0.83651630373780790f,0.22414386804185735f,-0.12940952255092145f,0,0,0,0,0,0,0,0,0,0,0,0,0,0,0,0},
  {0.33267055295095690f,0.80689150931333880f,0.45987750211933130f,-0.13501102001039084f,-0.08544127388224149f,0.03522629188210066f,0,0,0,0,0,0,0,0,0,0,0,0,0,0},
  {0.23037781330885523f,0.71484657055254150f,0.63088076792959040f,-0.02798376941698385f,-0.18703481171888114f,0.03084138183598697f,0.03288301166698295f,-0.01059740178499728f,0,0,0,0,0,0,0,0,0,0,0,0},
  {0.16010239797412501f,0.60382926979747290f,0.72430852843857440f,0.13842814590110342f,-0.24229488706619015f,-0.03224486958502952f,0.07757149384006515f,-0.00624149021301171f,-0.01258075199901553f,0.00333572528500155f,0,0,0,0,0,0,0,0,0,0},
  {0.11154074335008017f,0.49462389039838540f,0.75113390802157750f,0.31525035170924320f,-0.22626469396516913f,-0.12976686756709563f,0.09750160558707936f,0.02752286553001629f,-0.03158203931803116f,0.00055384220099380f,0.00477725751101065f,-0.00107730108499558f,0,0,0,0,0,0,0,0},
  {0.07785205408506236f,0.39653931948230575f,0.72913209084655510f,0.46978228740535860f,-0.14390600392910627f,-0.22403618499416572f,0.07130921926705004f,0.08061260915108307f,-0.03802993693503463f,-0.01657454163101562f,0.01255099855601378f,0.00042957797300470f,-0.00180164070399983f,0.00035371380000104f,0,0,0,0,0,0},
  {0.05441584224308161f,0.31287159091446590f,0.67563073629801280f,0.58535468365486910f,-0.01582910525602389f,-0.28401554296242810f,0.00047248457399797f,0.12874742662018600f,-0.01736930100202211f,-0.04408825393106472f,0.01398102791701552f,0.00874609404701566f,-0.00487035299301066f,-0.00039174037299598f,0.00067544940599856f,-0.00011747678400228f,0,0,0,0},
  {0.03807794736316728f,0.24383467463766728f,0.60482312367677860f,0.65728807803663890f,0.13319738582208895f,-0.29327378327258685f,-0.09684078322087904f,0.14854074933476008f,0.03072568147832287f,-0.06763282905952399f,0.00025094711499194f,0.02236166212351524f,-0.00472320475789483f,-0.00428150368190472f,0.00184764688296113f,0.00023038576399541f,-0.00025196318899818f,0.00003934731999503f,0,0},
  {0.02667005790095082f,0.18817680007762133f,0.52720118893091980f,0.68845903945259210f,0.28117234366042650f,-0.24984642432648865f,-0.19594627437659665f,0.12736934033574265f,0.09305736460380659f,-0.07139414716586077f,-0.02945753682194567f,0.03321267405893324f,0.00360655356698839f,-0.01073317548297960f,0.00139535174699408f,0.00199240529499085f,-0.00068585669500468f,-0.00011646685499439f,0.00009358867000109f,-0.00001326420300235f}
};

__device__ __forceinline__ float mishf(float x){
  float sp = (x > 20.f) ? x : log1pf(__expf(x));
  return x * tanhf(sp);
}

// CDNA5 async global->LDS copy (ASYNCcnt-tracked, no VGPR data). 16B per lane.
__device__ __forceinline__ void async_b128(const float* g, unsigned lds_off){
  asm volatile("global_load_async_to_lds_b128 %0, %1, off"
               :: "v"(lds_off), "v"(g) : "memory");
}
__device__ __forceinline__ void wait_async0(){
  asm volatile("s_wait_asynccnt 0x0" ::: "memory");
}
__device__ __forceinline__ unsigned lds_off_of(const void* p){
  // generic LDS pointer: addr[31:0] is the wave-relative LDS offset
  return (unsigned)(uintptr_t)p;
}

// -------- lift: h = fc1(fc0(concat(x, grid))) --------
__global__ void __launch_bounds__(256) lift_kernel(const float* __restrict__ x,
    const float* __restrict__ fc0w, const float* __restrict__ fc0b,
    const float* __restrict__ fc1w, const float* __restrict__ fc1b,
    float* __restrict__ h){
  __shared__ float sW0[WID*2], sB0[WID], sW1[WID*WID], sB1[WID];
  for (int i = threadIdx.x; i < WID*2; i += blockDim.x) sW0[i] = fc0w[i];
  for (int i = threadIdx.x; i < WID;   i += blockDim.x){ sB0[i] = fc0b[i]; sB1[i] = fc1b[i]; }
  for (int i = threadIdx.x; i < WID*WID; i += blockDim.x) sW1[i] = fc1w[i];
  __syncthreads();
  int g = blockIdx.x*blockDim.x + threadIdx.x;       // (b,n)
  int b = g >> 10, n = g & 1023;
  float v0 = x[g];
  float v1 = (float)n * (1.0f/1023.0f);
  float t[WID];
  #pragma unroll
  for (int c = 0; c < WID; ++c) t[c] = fmaf(sW0[c*2], v0, fmaf(sW0[c*2+1], v1, sB0[c]));
  for (int c = 0; c < WID; ++c){
    float s = sB1[c];
    #pragma unroll
    for (int k = 0; k < WID; ++k) s = fmaf(sW1[c*WID+k], t[k], s);
    h[(((size_t)(b*WID + c)) << 10) + n] = s;
  }
}

// -------- xf = mean over spatial --------
__global__ void xf_kernel(const float* __restrict__ h, float* __restrict__ xf){
  __shared__ float red[128];
  int row = blockIdx.x;                               // b*64+c
  const float* p = h + ((size_t)row << 10);
  float s = 0.f;
  for (int i = threadIdx.x; i < NSZ; i += 128) s += p[i];
  red[threadIdx.x] = s; __syncthreads();
  for (int off = 64; off > 0; off >>= 1){
    if (threadIdx.x < off) red[threadIdx.x] += red[threadIdx.x + off];
    __syncthreads();
  }
  if (threadIdx.x == 0) xf[row] = red[0] * (1.0f/NSZ);
}

// -------- gate: lam[hd][b][e] = softmax over experts --------
__global__ void gate_kernel(const float* __restrict__ label,
    const float* __restrict__ gwl, const float* __restrict__ gbl,
    const float* __restrict__ gwg, const float* __restrict__ gbg,
    const float* __restrict__ xf, float* __restrict__ lam){
  int t = threadIdx.x; if (t >= NHID*BSZ) return;
  int hd = t >> 5, b = t & 31;
  float ls = 0.f;
  #pragma unroll
  for (int l = 0; l < NLAB; ++l) ls += label[l];
  ls *= (1.0f/NLAB);
  float lf[NLIFT];
  #pragma unroll
  for (int o = 0; o < NLIFT; ++o){
    const float* wr = gwl + (size_t)(hd*NLIFT + o)*WID;
    float rs = 0.f;
    for (int c = 0; c < WID; ++c) rs += wr[c];
    lf[o] = rs*ls + gbl[hd*NLIFT + o];
  }
  float logits[NEXP]; float mx = -1e30f;
  #pragma unroll
  for (int e = 0; e < NEXP; ++e){
    const float* wr = gwg + (size_t)(hd*NEXP + e)*(WID+NLIFT);
    float s = gbg[hd*NEXP + e];
    for (int c = 0; c < WID;   ++c) s = fmaf(xf[b*WID + c], wr[c], s);
    for (int j = 0; j < NLIFT; ++j) s = fmaf(lf[j], wr[WID + j], s);
    logits[e] = s; mx = fmaxf(mx, s);
  }
  float den = 0.f;
  #pragma unroll
  for (int e = 0; e < NEXP; ++e){ logits[e] = __expf(logits[e]-mx); den += logits[e]; }
  float inv = 1.0f/den;
  #pragma unroll
  for (int e = 0; e < NEXP; ++e) lam[(hd*BSZ + b)*NEXP + e] = logits[e]*inv;
}

// -------- 4-level periodic DWT; d1..d3 -> pyramid, coarse a & d4 -> mode-major coefT --------
// coefT[(half*64 + x)*2048 + b*64 + i]   (half 0 = a, 1 = d4)
__global__ void __launch_bounds__(128) dwt_kernel(const float* __restrict__ h,
    float* __restrict__ dwt, float* __restrict__ coefT, int e){
  __shared__ __align__(16) float sA[NSZ];
  __shared__ float sB[NSZ/2], sD4[NMODES], ha[LMAXF], gg[LMAXF];
  int row = blockIdx.x;
  const float* src = h + ((size_t)row << 10);
  float* drow = dwt + ((size_t)row << 10);
  int L = 2*(e+1);
  if (threadIdx.x < LMAXF){
    int t = threadIdx.x;
    ha[t] = (t < L) ? DB_LO[e][t] : 0.f;
    gg[t] = (t < L) ? DB_LO[e][L-1-t] * ((t & 1) ? -1.f : 1.f) : 0.f;
  }
  {
    unsigned offA = lds_off_of(sA);
    for (int i = threadIdx.x*4; i < NSZ; i += 128*4) async_b128(src + i, offA + i*4);
    wait_async0();
  }
  __syncthreads();
  int n = NSZ;
  for (int lev = 0; lev < NLEV; ++lev){
    int half = n >> 1;
    for (int j = threadIdx.x; j < half; j += blockDim.x){
      float a = 0.f, d = 0.f;
      #pragma unroll
      for (int t = 0; t < LMAXF; ++t){
        int idx = 2*j + t; if (idx >= n) idx -= n;   // periodic wrap
        float v = sA[idx];
        a = fmaf(v, ha[t], a);
        d = fmaf(v, gg[t], d);
      }
      sB[j] = a;
      if (lev == NLEV-1) sD4[j] = d;
      else               drow[half + j] = d;          // detail offset == its length
    }
    __syncthreads();
    for (int j = threadIdx.x; j < half; j += blockDim.x) sA[j] = sB[j];
    __syncthreads();
    n = half;
  }
  for (int j = threadIdx.x; j < NMODES; j += blockDim.x){
    coefT[(size_t)j*2048 + row]             = sA[j];   // a panel, mode j
    coefT[(size_t)(NMODES + j)*2048 + row]  = sD4[j];  // d4 panel, mode j
  }
}

// -------- weight transpose for the mode-major GEMMs: wt[(half*64+x)][i][o] --------
__global__ void __launch_bounds__(256) wtrans_kernel(const float* __restrict__ w1,
    const float* __restrict__ w2, float* __restrict__ wt, int hd, int e){
  int g = blockIdx.x*256 + threadIdx.x;               // 0 .. 2*64*64*64-1
  int half = g >> 18;
  int rem  = g & 262143;
  int i = rem >> 12, o = (rem >> 6) & 63, x = rem & 63;
  const float* W = (half ? w2 : w1) + ((size_t)(hd*NEXP + e))*262144;
  wt[((size_t)(half*64 + x)*64 + i)*64 + o] = W[(size_t)(i*64 + o)*64 + x];
}

// -------- per-mode channel mix: async-stage A(8KB)+W(16KB) into LDS, fp32 WMMA from LDS --------
__global__ void __launch_bounds__(256) mix_kernel(const float* __restrict__ coefT,
    const float* __restrict__ wt, float* __restrict__ mix){
  __shared__ __align__(16) float sA[2048];            // [b][i]
  __shared__ __align__(16) float sW[4096];            // [i][o]
  int x = blockIdx.x, half = blockIdx.y;
  int p = half*NMODES + x;
  const float* gA = coefT + (size_t)p*2048;
  const float* gW = wt    + (size_t)p*4096;
  int tid = threadIdx.x;
  {
    unsigned offA = lds_off_of(sA), offW = lds_off_of(sW);
    for (int i = tid*4; i < 2048; i += 1024) async_b128(gA + i, offA + i*4);
    for (int i = tid*4; i < 4096; i += 1024) async_b128(gW + i, offW + i*4);
    wait_async0();
  }
  __syncthreads();
  int wave = tid >> 5, lane = tid & 31;
  int b0 = (wave >> 2)*16, o0 = (wave & 3)*16;        // 2x4 tiles of 16x16 -> 32x64
  int mr = lane & 15, hi = lane >> 4;
  v8f c = {};
  #pragma unroll
  for (int k0 = 0; k0 < WID; k0 += 4){
    int ka = k0 + hi*2;                               // A: 16x4 fp32, lanes 0-15 K={0,1}, 16-31 K={2,3}
    v2f A = *(const v2f*)&sA[(b0+mr)*WID + ka];       // ds_load_b64
    v2f Bv;
    Bv.x = sW[(k0 + hi    )*WID + (o0+mr)];
    Bv.y = sW[(k0 + hi + 2)*WID + (o0+mr)];
    c = __builtin_amdgcn_wmma_f32_16x16x4_f32(false, A, false, Bv, (short)0, c, false, false);
  }
  int off = half*64;
  #pragma unroll
  for (int r = 0; r < 8; ++r){                        // C/D: VGPR r -> M=r (lo lanes), M=r+8 (hi lanes)
    int mm = b0 + hi*8 + r;
    int oo = o0 + mr;
    mix[(size_t)(mm*WID + oo)*128 + off + x] = c[r];
  }
}

// -------- 4-level IDWT of one (b,c) row + lam-weighted accumulate --------
__global__ void __launch_bounds__(128) idwt_kernel(const float* __restrict__ dwt,
    const float* __restrict__ mix, const float* __restrict__ lam,
    float* __restrict__ acc, int hd, int e){
  __shared__ __align__(16) float sA[NSZ];
  __shared__ __align__(16) float sD[NSZ/2];
  __shared__ float sB[NSZ], h0[LMAXF], h1[LMAXF];
  int row = blockIdx.x; int b = row >> 6;
  int L = 2*(e+1);
  if (threadIdx.x < LMAXF){
    int t = threadIdx.x;
    int j = LMAXF - 1 - t;                            // synthesis = full-pad reverse of analysis
    h0[t] = (j < L) ? DB_LO[e][j] : 0.f;
    h1[t] = (j < L) ? DB_LO[e][L-1-j] * ((j & 1) ? -1.f : 1.f) : 0.f;
  }
  const float* mrow = mix + (size_t)row*128;
  {
    unsigned offA = lds_off_of(sA), offD = lds_off_of(sD);
    for (int i = threadIdx.x*4; i < 64; i += 128*4){  // threads 0..15
      async_b128(mrow + i,       offA + i*4);
      async_b128(mrow + 64 + i,  offD + i*4);
    }
    wait_async0();
  }
  __syncthreads();
  int k = 64;
  for (int lev = 0; lev < NLEV; ++lev){
    int n2 = 2*k;
    for (int nn = threadIdx.x; nn < n2; nn += blockDim.x){
      float r = 0.f;
      #pragma unroll
      for (int t = 0; t < LMAXF; ++t){
        int i = nn + t - (LMAXF-1);
        if (i >= 0 && i <= n2-2 && !(i & 1)){
          int p = i >> 1;
          r = fmaf(sA[p], h0[t], r);
          r = fmaf(sD[p], h1[t], r);
        }
      }
      if (nn < LMAXF-2){                              // wrap-add of conv tail
        int m2 = n2 + nn;
        #pragma unroll
        for (int t = 0; t < LMAXF; ++t){
          int i = m2 + t - (LMAXF-1);
          if (i >= 0 && i <= n2-2 && !(i & 1)){
            int p = i >> 1;
            r = fmaf(sA[p], h0[t], r);
            r = fmaf(sD[p], h1[t], r);
          }
        }
      }
      sB[nn] = r;
    }
    __syncthreads();
    for (int i = threadIdx.x; i < n2; i += blockDim.x) sA[i] = sB[i];
    if (lev < NLEV-1){
      const float* dsrc = dwt + ((size_t)row << 10) + n2;   // next detail at offset == its length
      unsigned offD = lds_off_of(sD);
      for (int i = threadIdx.x*4; i < n2; i += 128*4) async_b128(dsrc + i, offD + i*4);
      wait_async0();
    }
    __syncthreads();
    k = n2;
  }
  float w = lam[(hd*BSZ + b)*NEXP + e];
  float* arow = acc + ((size_t)row << 10);
  if (e == 0){
    for (int i = threadIdx.x; i < NSZ; i += blockDim.x) arow[i] = w * sA[i];
  } else {
    for (int i = threadIdx.x; i < NSZ; i += blockDim.x) arow[i] += w * sA[i];
  }
}

// -------- h = mish(acc + conv1x1(h, w_w, w_b)), in-place safe (each thread owns one n) --------
__global__ void __launch_bounds__(256) combine_kernel(float* __restrict__ h,
    const float* __restrict__ acc, const float* __restrict__ ww,
    const float* __restrict__ wb, int hd){
  __shared__ float sW[WID*WID], sBv[WID];
  const float* wsrc = ww + (size_t)hd*WID*WID;
  for (int i = threadIdx.x; i < WID*WID; i += blockDim.x) sW[i] = wsrc[i];
  for (int i = threadIdx.x; i < WID;     i += blockDim.x) sBv[i] = wb[hd*WID + i];
  __syncthreads();
  int b = blockIdx.y;
  int n = blockIdx.x*blockDim.x + threadIdx.x;
  float hv[WID];
  #pragma unroll
  for (int c = 0; c < WID; ++c) hv[c] = h[(((size_t)(b*WID + c)) << 10) + n];
  for (int c = 0; c < WID; ++c){
    float s = sBv[c];
    #pragma unroll
    for (int k = 0; k < WID; ++k) s = fmaf(sW[c*WID + k], hv[k], s);
    size_t idx = (((size_t)(b*WID + c)) << 10) + n;
    h[idx] = mishf(s + acc[idx]);
  }
}

// -------- fused head: out = fc3(mish(fc2(h))) --------
__global__ void __launch_bounds__(128) head_kernel(const float* __restrict__ h,
    const float* __restrict__ w2, const float* __restrict__ b2,
    const float* __restrict__ w3, const float* __restrict__ b3,
    float* __restrict__ out){
  __shared__ float sW[128*WID], sB[128], sW3[128];
  for (int i = threadIdx.x; i < 128*WID; i += blockDim.x) sW[i] = w2[i];
  for (int i = threadIdx.x; i < 128;     i += blockDim.x){ sB[i] = b2[i]; sW3[i] = w3[i]; }
  __syncthreads();
  int g = blockIdx.x*blockDim.x + threadIdx.x;
  int b = g >> 10, n = g & 1023;
  float hv[WID];
  #pragma unroll
  for (int c = 0; c < WID; ++c) hv[c] = h[(((size_t)(b*WID + c)) << 10) + n];
  float o = b3[0];
  for (int k = 0; k < 128; ++k){
    float s = sB[k];
    #pragma unroll
    for (int c = 0; c < WID; ++c) s = fmaf(sW[k*WID + c], hv[c], s);
    o = fmaf(sW3[k], mishf(s), o);
  }
  out[g] = o;
}

extern "C" void kernel_launch(void* const* d_in, const int* in_sizes, int n_in,
                              void* d_out, int out_size, void* d_ws, size_t ws_size,
                              hipStream_t stream){
  (void)in_sizes; (void)n_in; (void)out_size; (void)ws_size;
  const float* x     = (const float*)d_in[0];
  const float* label = (const float*)d_in[1];
  const float* fc0w  = (const float*)d_in[2];
  const float* fc0b  = (const float*)d_in[3];
  const float* fc1w  = (const float*)d_in[4];
  const float* fc1b  = (const float*)d_in[5];
  const float* gwl   = (const float*)d_in[6];
  const float* gbl   = (const float*)d_in[7];
  const float* gwg   = (const float*)d_in[8];
  const float* gbg   = (const float*)d_in[9];
  const float* ew1   = (const float*)d_in[10];
  const float* ew2   = (const float*)d_in[11];
  const float* ww    = (const float*)d_in[12];
  const float* wb    = (const float*)d_in[13];
  const float* fc2w  = (const float*)d_in[14];
  const float* fc2b  = (const float*)d_in[15];
  const float* fc3w  = (const float*)d_in[16];
  const float* fc3b  = (const float*)d_in[17];

  float* ws    = (float*)d_ws;
  float* h     = ws;                        // 2,097,152 f32
  float* acc   = h     + (size_t)2097152;   // 2,097,152
  float* dwtb  = acc   + (size_t)2097152;   // 2,097,152
  float* mixb  = dwtb  + (size_t)2097152;   //   262,144
  float* coefT = mixb  + (size_t)262144;    //   262,144  (128 mode-panels of 2048)
  float* wtb   = coefT + (size_t)262144;    //   524,288  (128 mode-panels of 4096)
  float* xfb   = wtb   + (size_t)524288;    //     2,048
  float* lamb  = xfb   + (size_t)2048;      //     1,280   (total ~29 MB)

  lift_kernel<<<(BSZ*NSZ)/256, 256, 0, stream>>>(x, fc0w, fc0b, fc1w, fc1b, h);
  xf_kernel  <<<BSZ*WID, 128, 0, stream>>>(h, xfb);
  gate_kernel<<<1, 128, 0, stream>>>(label, gwl, gbl, gwg, gbg, xfb, lamb);

  for (int hd = 0; hd < NHID; ++hd){
    for (int e = 0; e < NEXP; ++e){
      dwt_kernel  <<<BSZ*WID, 128, 0, stream>>>(h, dwtb, coefT, e);
      wtrans_kernel<<<2048, 256, 0, stream>>>(ew1, ew2, wtb, hd, e);
      mix_kernel  <<<dim3(NMODES, 2), 256, 0, stream>>>(coefT, wtb, mixb);
      idwt_kernel <<<BSZ*WID, 128, 0, stream>>>(dwtb, mixb, lamb, acc, hd, e);
    }
    combine_kernel<<<dim3(NSZ/256, BSZ), 256, 0, stream>>>(h, acc, ww, wb, hd);
  }

  head_kernel<<<(BSZ*NSZ)/128, 128, 0, stream>>>(h, fc2w, fc2b, fc3w, fc3b, (float*)d_out);
}